// Voxelize_52321291600272
// MI455X (gfx1250) — compile-verified
//
#include <hip/hip_runtime.h>
#include <hip/hip_bf16.h>

typedef float v2f __attribute__((ext_vector_type(2)));
typedef float v8f __attribute__((ext_vector_type(8)));

#define NB 8
#define NVERT 2048
#define NFAC 512
#define GW 96
#define VGRID (GW * GW * GW)            // 884736 points per batch
#define FTILES (NFAC / 4)               // 128 facet tiles (4 facets x 4 lam-rows = 16 rows)
#define A_FLOATS (FTILES * 64)          // 8192 floats = 32KB per batch (A fragments)
#define PTILES (VGRID / 16)             // 55296 point tiles per batch
#define TILES_PER_WAVE 8
#define WAVES_PER_BLOCK 8
#define TILES_PER_BLOCK (TILES_PER_WAVE * WAVES_PER_BLOCK)   // 64
#define BLOCKS_X (PTILES / TILES_PER_BLOCK)                  // 864

// ---------------------------------------------------------------------------
// Kernel 1: per-facet precompute.  lam_r(P) = M[r][0]*x + M[r][1]*y +
// M[r][2]*z + M[r][3], r=0..3 (row 3 encodes lam4 = 1 - sum).  Scattered
// straight into the V_WMMA_F32_16X16X4_F32 A-operand layout:
//   lane m (0..15)  : VGPR0 = A[m][K=0], VGPR1 = A[m][K=1]
//   lane m+16       : VGPR0 = A[m][K=2], VGPR1 = A[m][K=3]
// stored as ws[tile][lane][2] so the main kernel does one ds_load_b64/lane.
// ---------------------------------------------------------------------------
__global__ void vox_prep_kernel(const float* __restrict__ verts,
                                const int* __restrict__ facets,
                                float* __restrict__ wsA) {
    int gid = blockIdx.x * blockDim.x + threadIdx.x;
    if (gid >= NB * NFAC) return;
    int b = gid / NFAC;
    int f = gid - b * NFAC;

    const int* fi = facets + (size_t)(b * NFAC + f) * 4;
    float vx[4], vy[4], vz[4];
    #pragma unroll
    for (int j = 0; j < 4; ++j) {
        const float* p = verts + ((size_t)b * NVERT + (unsigned)fi[j]) * 3;
        vx[j] = p[0]; vy[j] = p[1]; vz[j] = p[2];
    }
    // Edge vectors e_j = v_j - v_3; T has e_j as columns, adj = adjugate(T).
    float a  = vx[0] - vx[3], bq = vx[1] - vx[3], c  = vx[2] - vx[3];
    float d  = vy[0] - vy[3], e  = vy[1] - vy[3], ff = vy[2] - vy[3];
    float g  = vz[0] - vz[3], h  = vz[1] - vz[3], i2 = vz[2] - vz[3];

    float adj[3][3] = {
        { e * i2 - ff * h,  c * h - bq * i2,  bq * ff - c * e },
        { ff * g - d * i2,  a * i2 - c * g,   c * d - a * ff  },
        { d * h - e * g,    bq * g - a * h,   a * e - bq * d  }
    };
    float det = a * (e * i2 - ff * h) - bq * (d * i2 - ff * g) + c * (d * h - e * g);
    float inv = 1.0f / det;

    float M[4][4];
    if (det != 0.0f && !isinf(inv)) {
        #pragma unroll
        for (int r = 0; r < 3; ++r) {
            M[r][0] = adj[r][0] * inv;
            M[r][1] = adj[r][1] * inv;
            M[r][2] = adj[r][2] * inv;
            M[r][3] = -(adj[r][0] * vx[3] + adj[r][1] * vy[3] + adj[r][2] * vz[3]) * inv;
        }
        #pragma unroll
        for (int cc = 0; cc < 3; ++cc)
            M[3][cc] = -(M[0][cc] + M[1][cc] + M[2][cc]);
        M[3][3] = 1.0f - (M[0][3] + M[1][3] + M[2][3]);
    } else {
        // Degenerate tet: lam0 = -1 everywhere -> never inside (reference's
        // NaN/inf comparisons also yield "not inside").
        #pragma unroll
        for (int r = 0; r < 4; ++r)
            #pragma unroll
            for (int cc = 0; cc < 4; ++cc) M[r][cc] = 0.0f;
        M[0][3] = -1.0f;
    }

    float* base = wsA + (size_t)b * A_FLOATS + (size_t)(f >> 2) * 64;
    int mrow = (f & 3) * 4;
    #pragma unroll
    for (int r = 0; r < 4; ++r) {
        #pragma unroll
        for (int cc = 0; cc < 4; ++cc) {
            int lane = mrow + r + ((cc >= 2) ? 16 : 0);
            base[lane * 2 + (cc & 1)] = M[r][cc];
        }
    }
}

// Reduce one 16x16 lam tile (2 facets per lane) into the running max of
// per-facet scores s = min(min(lam), 1 - max(lam)); inside <=> acc >= 0.
__device__ __forceinline__ void reduce_tile(const v8f& lam, float& acc) {
    float mn0 = fminf(fminf(lam[0], lam[1]), fminf(lam[2], lam[3]));
    float mx0 = fmaxf(fmaxf(lam[0], lam[1]), fmaxf(lam[2], lam[3]));
    float mn1 = fminf(fminf(lam[4], lam[5]), fminf(lam[6], lam[7]));
    float mx1 = fmaxf(fmaxf(lam[4], lam[5]), fmaxf(lam[6], lam[7]));
    float s0 = fminf(mn0, 1.0f - mx0);
    float s1 = fminf(mn1, 1.0f - mx1);
    acc = fmaxf(acc, fmaxf(s0, s1));
}

// ---------------------------------------------------------------------------
// Kernel 2: main voxelization.  One block = one batch slice; A fragments for
// the whole batch live in 32KB LDS.  Each wave sweeps PAIRS of 16-point
// z-runs: one ds_load_b64 of an A fragment feeds TWO wmmas (two B fragments),
// halving LDS traffic per matrix op.  B fragment is (x,y | z_n,1); each
// 16x16 lam tile reduces to max_f min(min(lam), 1 - max(lam)).
// ---------------------------------------------------------------------------
__global__ void __launch_bounds__(256)
vox_main_kernel(const float* __restrict__ wsA, float* __restrict__ out) {
    __shared__ float As[A_FLOATS];

    const int b = blockIdx.y;
    const float* src = wsA + (size_t)b * A_FLOATS;
    for (int idx = threadIdx.x; idx < A_FLOATS; idx += 256)
        As[idx] = src[idx];
    __syncthreads();

    const int lane = threadIdx.x & 31;
    const int wave = threadIdx.x >> 5;
    const int tile0 = blockIdx.x * TILES_PER_BLOCK + wave * TILES_PER_WAVE;
    const v2f* Af = (const v2f*)As;   // [FTILES][32] pairs

    for (int it = 0; it < TILES_PER_WAVE; it += 2) {
        const int pt = tile0 + it;        // pair covers points v0 .. v0+31
        const int v0 = pt * 16;
        const int gi = v0 / (GW * GW);
        const int gj = (v0 / GW) % GW;
        const int k0 = v0 % GW;           // 32 | GW, so i,j fixed across the pair

        const float x  = (2.0f * gi + 1.0f - (float)GW) * (1.0f / GW);
        const float y  = (2.0f * gj + 1.0f - (float)GW) * (1.0f / GW);
        const float z0 = (2.0f * (k0 + (lane & 15)) + 1.0f - (float)GW) * (1.0f / GW);
        const float z1 = z0 + 32.0f * (1.0f / GW);

        // B rows: K0=x K1=y (lanes 0-15), K2=z K3=1 (lanes 16-31)
        v2f b0, b1;
        b0.x = (lane < 16) ? x : z0;
        b0.y = (lane < 16) ? y : 1.0f;
        b1.x = (lane < 16) ? x : z1;
        b1.y = b0.y;

        float acc0 = -1.0f, acc1 = -1.0f;
        #pragma unroll 4
        for (int t = 0; t < FTILES; ++t) {
            v2f afrag = Af[t * 32 + lane];
            v8f cz = {0.f, 0.f, 0.f, 0.f, 0.f, 0.f, 0.f, 0.f};
            v8f lamA = __builtin_amdgcn_wmma_f32_16x16x4_f32(
                false, afrag, false, b0, (short)0, cz, false, false);
            v8f lamB = __builtin_amdgcn_wmma_f32_16x16x4_f32(
                false, afrag, false, b1, (short)0, cz, false, false);
            reduce_tile(lamA, acc0);
            reduce_tile(lamB, acc1);
        }

        // lanes 0-15 hold facets {4t,4t+1}; lanes 16-31 hold {4t+2,4t+3}
        // for the same 16 points -> combine the two half-waves.
        acc0 = fmaxf(acc0, __shfl_xor(acc0, 16, 32));
        acc1 = fmaxf(acc1, __shfl_xor(acc1, 16, 32));

        if (lane < 16) {
            float* dst = out + (size_t)b * VGRID + v0 + lane;
            dst[0]  = (acc0 >= 0.0f) ? 1.0f : 0.0f;
            dst[16] = (acc1 >= 0.0f) ? 1.0f : 0.0f;
        }
    }
}

extern "C" void kernel_launch(void* const* d_in, const int* in_sizes, int n_in,
                              void* d_out, int out_size, void* d_ws, size_t ws_size,
                              hipStream_t stream) {
    const float* verts  = (const float*)d_in[0];
    const int*   facets = (const int*)d_in[1];
    float*       out    = (float*)d_out;
    float*       wsA    = (float*)d_ws;   // needs NB*A_FLOATS*4 = 256KB

    vox_prep_kernel<<<dim3((NB * NFAC + 255) / 256), 256, 0, stream>>>(verts, facets, wsA);
    vox_main_kernel<<<dim3(BLOCKS_X, NB), 256, 0, stream>>>(wsA, out);
}